// DecodeSPM_6708738916373
// MI455X (gfx1250) — compile-verified
//
#include <hip/hip_runtime.h>
#include <math.h>
#include <stdint.h>

// ---------------- DecodeSPM for gfx1250 (wave32) ----------------
// x: (1, 35, 1024, 1024) f32. Outputs (flat concat):
//   root [256,2] | kp [256,17,2] | valids [256]  -> 9472 floats.

#define SDIM      1024
#define SS        (SDIM * SDIM)
#define NKP       17
#define MAX_ROOTS 256
#define RADIUS    10        // DIST_THR = 10.0 -> d2 <= 100 (integer lattice)
#define DIST2     100
#define NEG_INF   (-__builtin_inff())

// argmax with reference tie-break: jnp.argmax picks the LOWEST flat index.
__device__ __forceinline__ void argmaxPair(float& v, int& i, float ov, int oi) {
  if (ov > v || (ov == v && oi < i)) { v = ov; i = oi; }
}

// wave32 butterfly argmax; all lanes end with the winner.
__device__ __forceinline__ void waveArgmax(float& v, int& i) {
#pragma unroll
  for (int off = 16; off > 0; off >>= 1) {
    float ov = __shfl_xor(v, off, 32);
    int   oi = __shfl_xor(i, off, 32);
    argmaxPair(v, i, ov, oi);
  }
}

// ---------------- Kernel A: scores + per-tile argmax -------------------
// One block per 32x32 tile; 256 threads x 4 consecutive pixels (b128 loads).
__global__ void __launch_bounds__(256) spm_scores_tiles(
    const float* __restrict__ x, float* __restrict__ scores,
    float* __restrict__ tileVal, int* __restrict__ tileIdx) {
  const int tile = blockIdx.x;               // 0..1023
  const int ty = tile >> 5, tx = tile & 31;
  const int t  = threadIdx.x;                // 0..255
  const int r  = t >> 3;                     // row in tile 0..31
  const int c  = (t & 7) << 2;               // col in tile 0,4,..,28
  const int y  = (ty << 5) + r;
  const int x0 = (tx << 5) + c;
  const int base = y * SDIM + x0;

  const float4 q = *(const float4*)(x + base);          // channel-0, b128
  __builtin_prefetch(x + base + 8 * SDIM, 0, 0);        // global_prefetch_b8

  float vv[4] = {q.x, q.y, q.z, q.w};
  float4 sc;
  float* scp = (float*)&sc;
  float bv = NEG_INF;
  int   bi = base;
#pragma unroll
  for (int j = 0; j < 4; ++j) {
    float conf = 1.0f / (1.0f + expf(-vv[j]));          // sigmoid
    float s = (conf > 0.8f) ? conf : NEG_INF;           // CONF_THR
    scp[j] = s;
    if (s > bv) { bv = s; bi = base + j; }              // strict > keeps lowest idx
  }
  *(float4*)(scores + base) = sc;                       // b128 store

  __shared__ float sv[256];
  __shared__ int   si[256];
  sv[t] = bv; si[t] = bi;
  __syncthreads();
  for (int stp = 128; stp > 0; stp >>= 1) {
    if (t < stp) argmaxPair(sv[t], si[t], sv[t + stp], si[t + stp]);
    __syncthreads();
  }
  if (t == 0) { tileVal[tile] = sv[0]; tileIdx[tile] = si[0]; }
}

// ---------------- Kernel B: greedy NMS, single workgroup ---------------
// LDS max tree: 1024 tile (val,idx) + 32 per-tile-row super (val,idx).
__global__ void __launch_bounds__(1024, 1) spm_nms(
    float* __restrict__ scores, const float* __restrict__ tileValG,
    const int* __restrict__ tileIdxG, int* __restrict__ rootIdx,
    int* __restrict__ rootValid) {
  __shared__ float tval[1024];
  __shared__ int   tidx[1024];
  __shared__ float sval[32];
  __shared__ int   sidx[32];
  __shared__ int   bcIdx, bcValid;

  const int t    = threadIdx.x;
  const int lane = t & 31;
  const int wave = t >> 5;

  tval[t] = tileValG[t];
  tidx[t] = tileIdxG[t];
  __syncthreads();
  {  // build super maxes: wave w reduces tile row w (32 tiles)
    float v  = tval[(wave << 5) + lane];
    int   id = tidx[(wave << 5) + lane];
    waveArgmax(v, id);
    if (lane == 0) { sval[wave] = v; sidx[wave] = id; }
  }
  __syncthreads();

  for (int r = 0; r < MAX_ROOTS; ++r) {
    if (wave == 0) {  // global argmax over 32 super entries
      float v  = sval[lane];
      int   id = sidx[lane];
      waveArgmax(v, id);
      if (lane == 0) {
        bcIdx   = id;
        bcValid = (v != NEG_INF) ? 1 : 0;   // isfinite (scores in {(0.8,1), -inf})
        rootIdx[r]   = id;
        rootValid[r] = bcValid;
      }
    }
    __syncthreads();
    const int cidx  = bcIdx;
    const int valid = bcValid;

    if (!valid) {  // all remaining are -inf forever: fill tail, exit
      for (int rr = r + t; rr < MAX_ROOTS; rr += 1024) {
        rootIdx[rr] = 0;
        rootValid[rr] = 0;
      }
      break;
    }

    const int cy = cidx >> 10, cx = cidx & (SDIM - 1);

    // suppress disk of radius 10 (21x21 box, 441 candidate lattice points)
    if (t < 441) {
      int dy = t / 21 - RADIUS, dx = t % 21 - RADIUS;
      if (dy * dy + dx * dx <= DIST2) {
        int yy = cy + dy, xx = cx + dx;
        if ((unsigned)yy < SDIM && (unsigned)xx < SDIM)
          scores[yy * SDIM + xx] = NEG_INF;
      }
    }
    __threadfence_block();
    asm volatile("s_wait_storecnt 0" ::: "memory");  // gfx1250 split counters
    __syncthreads();

    // rescan affected tiles (<=4): one wave per tile, one lane per row
    const int ty0 = max(cy - RADIUS, 0) >> 5, ty1 = min(cy + RADIUS, SDIM - 1) >> 5;
    const int tx0 = max(cx - RADIUS, 0) >> 5, tx1 = min(cx + RADIUS, SDIM - 1) >> 5;
    const int nty = ty1 - ty0 + 1, ntx = tx1 - tx0 + 1;
    const int ntiles = nty * ntx;
    if (wave < ntiles) {
      const int tty = ty0 + wave / ntx, ttx = tx0 + wave % ntx;
      const int y   = (tty << 5) + lane;
      const int rowBase = y * SDIM + (ttx << 5);
      const float* rowp = scores + rowBase;
      float bv = NEG_INF;
      int   bi = rowBase;
#pragma unroll
      for (int k = 0; k < 8; ++k) {
        float4 q = *(const float4*)(rowp + (k << 2));   // b128
        float qa[4] = {q.x, q.y, q.z, q.w};
#pragma unroll
        for (int j = 0; j < 4; ++j) {
          if (qa[j] > bv) { bv = qa[j]; bi = rowBase + (k << 2) + j; }
        }
      }
      waveArgmax(bv, bi);
      if (lane == 0) { tval[(tty << 5) + ttx] = bv; tidx[(tty << 5) + ttx] = bi; }
    }
    __syncthreads();

    // refresh affected super rows (<=2)
    if (wave < nty) {
      const int row = ty0 + wave;
      float v  = tval[(row << 5) + lane];
      int   id = tidx[(row << 5) + lane];
      waveArgmax(v, id);
      if (lane == 0) { sval[row] = v; sidx[row] = id; }
    }
    __syncthreads();
  }
}

// ---------------- Kernel C: gather disp @ roots via ASYNC->LDS ----------
// 34 strided fp32 gathers per root land directly in LDS through the gfx1250
// async path (GLOBAL_LOAD_ASYNC_TO_LDS_B32, tracked by ASYNCcnt), then tanh
// is applied in place. Only 8.7k tanh total instead of the reference's 34M.
__global__ void __launch_bounds__(64) spm_finalize(
    const float* __restrict__ x, const int* __restrict__ rootIdx,
    const int* __restrict__ rootValid, float* __restrict__ out) {
  const int b = blockIdx.x;    // root id 0..255
  const int t = threadIdx.x;   // 0..63
  __shared__ float dv[34];

  const int idx   = rootIdx[b];
  const int valid = rootValid[b];
  const int cy = idx >> 10, cx = idx & (SDIM - 1);

  if (valid) {                 // uniform per block
    if (t < 34) {
      // LDS aperture keeps the wave-relative byte offset in addr[31:0].
      unsigned ldsoff = (unsigned)(uintptr_t)(&dv[t]);
      const float* gp = x + (size_t)(1 + t) * SS + (size_t)cy * SDIM + cx;
      asm volatile("global_load_async_to_lds_b32 %0, %1, off"
                   :: "v"(ldsoff), "v"(gp)
                   : "memory");
    }
    asm volatile("s_wait_asynccnt 0" ::: "memory");   // async gather done
    if (t < 34) dv[t] = tanhf(dv[t]);
  } else {
    if (t < 34) dv[t] = 0.0f;
  }
  __syncthreads();

  float* outRoot = out;
  float* outKp   = out + 2 * MAX_ROOTS;
  float* outVal  = out + 2 * MAX_ROOTS + MAX_ROOTS * NKP * 2;
  const float zc    = (float)(1.4142135623730951 * 1024.0);  // sqrt(2)*S
  const float scale = 4.0f;                                  // 4096/1024

  if (t < NKP) {
    float kx = 0.0f, ky = 0.0f;
    if (valid) {
      float ddx = dv[2 * t] * zc;
      float ddy = dv[2 * t + 1] * zc;
      float d = sqrtf(ddx * ddx + ddy * ddy);
      if (d >= 2.0f) {  // where(d < 2, 0, kp)
        kx = (ddx + (float)cx) * scale;
        ky = (ddy + (float)cy) * scale;
      }
    }
    outKp[(b * NKP + t) * 2 + 0] = kx;
    outKp[(b * NKP + t) * 2 + 1] = ky;
  }
  if (t == 0) {
    outRoot[2 * b + 0] = valid ? (float)cx * scale : 0.0f;
    outRoot[2 * b + 1] = valid ? (float)cy * scale : 0.0f;
    outVal[b] = valid ? 1.0f : 0.0f;
  }
}

// ---------------- launch ----------------
extern "C" void kernel_launch(void* const* d_in, const int* in_sizes, int n_in,
                              void* d_out, int out_size, void* d_ws, size_t ws_size,
                              hipStream_t stream) {
  (void)in_sizes; (void)n_in; (void)out_size; (void)ws_size;
  const float* x = (const float*)d_in[0];
  char* ws = (char*)d_ws;

  float* scores    = (float*)(ws);                               // 4 MB score grid
  float* tileVal   = (float*)(ws + (size_t)SS * 4);              // 4 KB
  int*   tileIdx   = (int*)  (ws + (size_t)SS * 4 + 4096);       // 4 KB
  int*   rootIdx   = (int*)  (ws + (size_t)SS * 4 + 8192);       // 1 KB
  int*   rootValid = (int*)  (ws + (size_t)SS * 4 + 8192 + 1024);// 1 KB
  float* out = (float*)d_out;

  spm_scores_tiles<<<SDIM, 256, 0, stream>>>(x, scores, tileVal, tileIdx);
  spm_nms<<<1, 1024, 0, stream>>>(scores, tileVal, tileIdx, rootIdx, rootValid);
  spm_finalize<<<MAX_ROOTS, 64, 0, stream>>>(x, rootIdx, rootValid, out);
}